// GAT_33363305955882
// MI455X (gfx1250) — compile-verified
//
#include <hip/hip_runtime.h>

// ---------------------------------------------------------------------------
// GATv2 (2 layers, heads=1) for MI455X / gfx1250, wave32.
//  * GEMMs: V_WMMA_F32_16X16X4_F32 (full fp32 precision; GEMMs here are
//    memory-bound so the f32 matrix pipe costs nothing vs f16), K templated
//    so the WMMA loop fully unrolls.
//  * X tiles staged via GLOBAL_LOAD_ASYNC_TO_LDS_B128 (ASYNCcnt) -- the
//    CDNA5 async copy engine feeds LDS directly, no VGPR round trip.
//  * Edge phase: wave-per-edge (H=64) / thread-per-edge (H=2) with
//    L2-resident node features (xl/xr/h ~38MB << 192MB L2), ordered-uint
//    atomicMax for segment-max, f32 atomics for segment-sum.
// ---------------------------------------------------------------------------

typedef float v2f __attribute__((ext_vector_type(2)));
typedef float v8f __attribute__((ext_vector_type(8)));

#define NEG_SLOPE 0.2f
#define GAT_EPS   1e-16f
// ford(-inf): order-preserving uint encoding of -infinity
#define FORD_NEG_INF 0x007FFFFFu

__device__ __forceinline__ unsigned ford(float f) {
  unsigned u = __float_as_uint(f);
  return (u & 0x80000000u) ? ~u : (u | 0x80000000u);
}
__device__ __forceinline__ float ford_inv(unsigned o) {
  unsigned u = (o & 0x80000000u) ? (o ^ 0x80000000u) : ~o;
  return __uint_as_float(u);
}
__device__ __forceinline__ float lrelu(float v) {
  return (v >= 0.f) ? v : NEG_SLOPE * v;
}

// ---------------------------------------------------------------------------
// Async-stage `rows` full rows of X (row-major [N,K], K%4==0) starting at
// row0 into LDS (row-major rows x K). The tile is contiguous in X, so we copy
// 16B chunks with GLOBAL_LOAD_ASYNC_TO_LDS_B128; rows past N are zero-filled
// with plain LDS stores. Caller's lds must be 16B aligned.
// Generic->LDS address: low 32 bits of a generic LDS pointer are the
// wave-relative LDS byte address (ISA 10.2 aperture rules), which is exactly
// what the async instruction's VDST consumes (added to the wave's LDS base).
// ---------------------------------------------------------------------------
__device__ __forceinline__ void gat_stage_tile_async(const float* __restrict__ X,
                                                     float* lds, int row0, int rows,
                                                     int N, int K) {
  const int kf4    = K >> 2;                      // float4 chunks per row
  const int total  = rows * kf4;
  const int vrows  = (N - row0) < rows ? (N - row0) : rows;
  const int nvalid = vrows * kf4;
  unsigned lbase = (unsigned)(uintptr_t)lds;
  unsigned long long gbase = (unsigned long long)(uintptr_t)(X + (size_t)row0 * K);
  for (int c = threadIdx.x; c < total; c += blockDim.x) {
    if (c < nvalid) {
      unsigned           lvds = lbase + ((unsigned)c << 4);
      unsigned long long gptr = gbase + ((unsigned long long)c << 4);
      asm volatile("global_load_async_to_lds_b128 %0, %1, off"
                   :: "v"(lvds), "v"(gptr) : "memory");
    } else {
      *(float4*)((char*)lds + ((size_t)c << 4)) = make_float4(0.f, 0.f, 0.f, 0.f);
    }
  }
  asm volatile("s_wait_asynccnt 0x0" ::: "memory");
  __syncthreads();
}

// ---------------------------------------------------------------------------
// init: acc[i*H+f] = bias[f]  (segment_sum + bias fused via accumulator seed)
//       mord[i] = ford(-inf), denom[i] = 0
// ---------------------------------------------------------------------------
__global__ void gat_init_accum(float* __restrict__ acc, const float* __restrict__ bias,
                               unsigned* __restrict__ mord, float* __restrict__ denom,
                               int N, int H) {
  int i = blockIdx.x * blockDim.x + threadIdx.x;
  if (i < N * H) acc[i] = bias[i % H];
  if (i < N) { mord[i] = FORD_NEG_INF; denom[i] = 0.f; }
}

// ---------------------------------------------------------------------------
// Dual GEMM: xl = X@Wl, xr = X@Wr.  X:[N,K] f32, W:[K,F] f32, F%16==0, K%4==0,
// F<=64. Block = 256 thr = 8 waves; wave -> (sel, ntile). 16 rows of X staged
// in LDS via async copy. f32 WMMA operand layouts per CDNA5 ISA 7.12.2:
//   A 16x4 : lanes 0-15 rows M=lane  hold K=k0,k0+1 ; lanes16-31 K=k0+2,k0+3
//   B 4x16 : lanes 0-15 cols N=lane  hold K=k0,k0+1 ; lanes16-31 K=k0+2,k0+3
//   C/D    : VGPR r -> M = r (+8 for upper half-lanes), N = lane&15
// KT: compile-time K (0 = runtime) so the WMMA loop fully unrolls.
// ---------------------------------------------------------------------------
template <int KT>
__global__ void __launch_bounds__(256)
gat_gemm_dual_wmma(const float* __restrict__ X,
                   const float* __restrict__ Wl, const float* __restrict__ Wr,
                   float* __restrict__ xl, float* __restrict__ xr,
                   int N, int Krt, int F) {
  extern __shared__ __align__(16) float lds[];   // 16 * K floats
  const int K = KT ? KT : Krt;
  int row0 = blockIdx.x * 16;
  gat_stage_tile_async(X, lds, row0, 16, N, K);

  int wave = threadIdx.x >> 5;
  int lane = threadIdx.x & 31;
  int half = lane >> 4;
  int m    = lane & 15;
  int ntiles = F >> 4;
  int sel = wave / ntiles;
  int nt  = wave - sel * ntiles;
  if (sel >= 2) return;
  const float* __restrict__ W = sel ? Wr : Wl;
  float* __restrict__ Y = sel ? xr : xl;
  int n = nt * 16 + m;

  v8f acc = {};
#pragma unroll
  for (int kk = 0; kk < K; kk += 4) {
    int k0 = kk + half * 2;
    v2f a, b;
    a.x = lds[m * K + k0];
    a.y = lds[m * K + k0 + 1];
    b.x = W[(size_t)k0 * F + n];
    b.y = W[(size_t)(k0 + 1) * F + n];
    acc = __builtin_amdgcn_wmma_f32_16x16x4_f32(false, a, false, b,
                                                (short)0, acc, false, false);
  }
#pragma unroll
  for (int r = 0; r < 8; ++r) {
    int row = row0 + half * 8 + r;
    if (row < N) Y[(size_t)row * F + n] = acc[r];
  }
}

// ---------------------------------------------------------------------------
// Layer-2 GEMM: K=H (%4==0), F small (<16) -> pad B columns with zero inside
// one 16-wide WMMA tile. Block handles 64 rows: waves -> (row tile, sel).
// ---------------------------------------------------------------------------
template <int KT>
__global__ void __launch_bounds__(256)
gat_gemm_dual_small_wmma(const float* __restrict__ X,
                         const float* __restrict__ Wl, const float* __restrict__ Wr,
                         float* __restrict__ xl, float* __restrict__ xr,
                         int N, int Krt, int F) {
  extern __shared__ __align__(16) float lds[];   // 64 * K floats
  const int K = KT ? KT : Krt;
  int rowB = blockIdx.x * 64;
  gat_stage_tile_async(X, lds, rowB, 64, N, K);

  int wave = threadIdx.x >> 5;
  int lane = threadIdx.x & 31;
  int half = lane >> 4;
  int m    = lane & 15;
  int mt  = wave >> 1;                           // 0..3 row tile
  int sel = wave & 1;                            // 0: Wl, 1: Wr
  const float* __restrict__ W = sel ? Wr : Wl;
  float* __restrict__ Y = sel ? xr : xl;
  int lrow0 = mt * 16;

  v8f acc = {};
#pragma unroll
  for (int kk = 0; kk < K; kk += 4) {
    int k0 = kk + half * 2;
    v2f a, b;
    a.x = lds[(lrow0 + m) * K + k0];
    a.y = lds[(lrow0 + m) * K + k0 + 1];
    b.x = (m < F) ? W[(size_t)k0 * F + m] : 0.f;
    b.y = (m < F) ? W[(size_t)(k0 + 1) * F + m] : 0.f;
    acc = __builtin_amdgcn_wmma_f32_16x16x4_f32(false, a, false, b,
                                                (short)0, acc, false, false);
  }
  if (m < F) {
#pragma unroll
    for (int r = 0; r < 8; ++r) {
      int row = rowB + lrow0 + half * 8 + r;
      if (row < N) Y[(size_t)row * F + m] = acc[r];
    }
  }
}

// ---------------------------------------------------------------------------
// Edge logits: e = att . LeakyReLU(xl[src] + xr[dst]); wave per edge,
// shuffle-reduce, ordered atomicMax for segment max. Self-loops are the tail
// [E, Etot): src = dst = idx - E.
// ---------------------------------------------------------------------------
__global__ void __launch_bounds__(256)
gat_edge_logits(const float* __restrict__ xl, const float* __restrict__ xr,
                const float* __restrict__ att,
                const int* __restrict__ srcE, const int* __restrict__ dstE,
                float* __restrict__ ebuf, unsigned* __restrict__ mord,
                int E, int Etot, int H) {
  int widx = (blockIdx.x * blockDim.x + threadIdx.x) >> 5;
  int lane = threadIdx.x & 31;
  if (widx >= Etot) return;
  int s = (widx < E) ? srcE[widx] : (widx - E);
  int d = (widx < E) ? dstE[widx] : (widx - E);
  float acc = 0.f;
  for (int f = lane; f < H; f += 32)
    acc += lrelu(xl[(size_t)s * H + f] + xr[(size_t)d * H + f]) * att[f];
#pragma unroll
  for (int off = 16; off; off >>= 1) acc += __shfl_xor(acc, off, 32);
  if (lane == 0) {
    ebuf[widx] = acc;
    atomicMax(mord + d, ford(acc));
  }
}

// Thread-per-edge variant for small H (layer 2: H==2).
__global__ void __launch_bounds__(256)
gat_edge_logits_small(const float* __restrict__ xl, const float* __restrict__ xr,
                      const float* __restrict__ att,
                      const int* __restrict__ srcE, const int* __restrict__ dstE,
                      float* __restrict__ ebuf, unsigned* __restrict__ mord,
                      int E, int Etot, int H) {
  int i = blockIdx.x * blockDim.x + threadIdx.x;
  if (i >= Etot) return;
  int s = (i < E) ? srcE[i] : (i - E);
  int d = (i < E) ? dstE[i] : (i - E);
  float acc = 0.f;
  for (int f = 0; f < H; ++f)
    acc += lrelu(xl[(size_t)s * H + f] + xr[(size_t)d * H + f]) * att[f];
  ebuf[i] = acc;
  atomicMax(mord + d, ford(acc));
}

// ---------------------------------------------------------------------------
// ex = exp(e - m[dst]); denom[dst] += ex  (ebuf overwritten in place)
// ---------------------------------------------------------------------------
__global__ void __launch_bounds__(256)
gat_edge_softmax_denom(const int* __restrict__ dstE,
                       float* __restrict__ ebuf, const unsigned* __restrict__ mord,
                       float* __restrict__ denom, int E, int Etot) {
  int i = blockIdx.x * blockDim.x + threadIdx.x;
  if (i >= Etot) return;
  int d = (i < E) ? dstE[i] : (i - E);
  float ex = __expf(ebuf[i] - ford_inv(mord[d]));
  ebuf[i] = ex;
  atomicAdd(denom + d, ex);
}

// ---------------------------------------------------------------------------
// out[dst] += (ex / (denom[dst]+eps)) * xl[src]; wave per edge, f32 atomics.
// ---------------------------------------------------------------------------
__global__ void __launch_bounds__(256)
gat_edge_aggregate(const float* __restrict__ xl,
                   const int* __restrict__ srcE, const int* __restrict__ dstE,
                   const float* __restrict__ ebuf, const float* __restrict__ denom,
                   float* __restrict__ out, int E, int Etot, int H) {
  int widx = (blockIdx.x * blockDim.x + threadIdx.x) >> 5;
  int lane = threadIdx.x & 31;
  if (widx >= Etot) return;
  int s = (widx < E) ? srcE[widx] : (widx - E);
  int d = (widx < E) ? dstE[widx] : (widx - E);
  float alpha = ebuf[widx] / (denom[d] + GAT_EPS);
  for (int f = lane; f < H; f += 32)
    atomicAdd(&out[(size_t)d * H + f], alpha * xl[(size_t)s * H + f]);
}

// Thread-per-edge variant for small H (layer 2: H==2).
__global__ void __launch_bounds__(256)
gat_edge_aggregate_small(const float* __restrict__ xl,
                         const int* __restrict__ srcE, const int* __restrict__ dstE,
                         const float* __restrict__ ebuf, const float* __restrict__ denom,
                         float* __restrict__ out, int E, int Etot, int H) {
  int i = blockIdx.x * blockDim.x + threadIdx.x;
  if (i >= Etot) return;
  int s = (i < E) ? srcE[i] : (i - E);
  int d = (i < E) ? dstE[i] : (i - E);
  float alpha = ebuf[i] / (denom[d] + GAT_EPS);
  for (int f = 0; f < H; ++f)
    atomicAdd(&out[(size_t)d * H + f], alpha * xl[(size_t)s * H + f]);
}

// ---------------------------------------------------------------------------
// host side
// ---------------------------------------------------------------------------
static inline int ceil_div(long long a, long long b) { return (int)((a + b - 1) / b); }

extern "C" void kernel_launch(void* const* d_in, const int* in_sizes, int /*n_in*/,
                              void* d_out, int /*out_size*/, void* d_ws, size_t /*ws_size*/,
                              hipStream_t stream) {
  // inputs: x, edge_index, Wl1, Wr1, att1, b1, Wl2, Wr2, att2, b2
  const float* x    = (const float*)d_in[0];
  const int*   ei   = (const int*)d_in[1];
  const float* Wl1  = (const float*)d_in[2];
  const float* Wr1  = (const float*)d_in[3];
  const float* att1 = (const float*)d_in[4];
  const float* b1   = (const float*)d_in[5];
  const float* Wl2  = (const float*)d_in[6];
  const float* Wr2  = (const float*)d_in[7];
  const float* att2 = (const float*)d_in[8];
  const float* b2   = (const float*)d_in[9];

  const int H    = in_sizes[5];            // 64
  const int Fin  = in_sizes[2] / H;        // 128
  const int N    = in_sizes[0] / Fin;      // 50000
  const int E    = in_sizes[1] / 2;        // 800000
  const int Fout = in_sizes[9];            // 2
  const int Etot = E + N;                  // self-loops appended

  const int* srcE = ei;
  const int* dstE = ei + E;

  // workspace carve-up (layer-2 buffers reuse layer-1 space; stream order
  // serializes kernels so reuse is safe)
  char* ws = (char*)d_ws;
  float*    xl1    = (float*)ws;                 ws += (size_t)N * H * sizeof(float);
  float*    xr1    = (float*)ws;                 ws += (size_t)N * H * sizeof(float);
  float*    h1     = (float*)ws;                 ws += (size_t)N * H * sizeof(float);
  float*    ebuf   = (float*)ws;                 ws += (size_t)Etot * sizeof(float);
  unsigned* mord   = (unsigned*)ws;              ws += (size_t)N * sizeof(unsigned);
  float*    denom  = (float*)ws;                 ws += (size_t)N * sizeof(float);
  // layer 2 node features overlay the (dead) xl1 region
  float* xl2 = xl1;
  float* xr2 = xl1 + (size_t)N * Fout;
  float* out = (float*)d_out;

  const int TPB = 256;

  // ---------------- layer 1 ----------------
  gat_init_accum<<<ceil_div((long long)N * H, TPB), TPB, 0, stream>>>(h1, b1, mord, denom, N, H);

  {
    size_t shmem = (size_t)16 * Fin * sizeof(float);
    if (Fin == 128)
      gat_gemm_dual_wmma<128><<<ceil_div(N, 16), TPB, shmem, stream>>>(
          x, Wl1, Wr1, xl1, xr1, N, Fin, H);
    else
      gat_gemm_dual_wmma<0><<<ceil_div(N, 16), TPB, shmem, stream>>>(
          x, Wl1, Wr1, xl1, xr1, N, Fin, H);
  }

  gat_edge_logits<<<ceil_div((long long)Etot * 32, TPB), TPB, 0, stream>>>(
      xl1, xr1, att1, srcE, dstE, ebuf, mord, E, Etot, H);

  gat_edge_softmax_denom<<<ceil_div(Etot, TPB), TPB, 0, stream>>>(
      dstE, ebuf, mord, denom, E, Etot);

  gat_edge_aggregate<<<ceil_div((long long)Etot * 32, TPB), TPB, 0, stream>>>(
      xl1, srcE, dstE, ebuf, denom, h1, E, Etot, H);

  // ---------------- layer 2 ----------------
  gat_init_accum<<<ceil_div((long long)N * Fout, TPB), TPB, 0, stream>>>(out, b2, mord, denom, N, Fout);

  {
    size_t shmem = (size_t)64 * H * sizeof(float);
    if (H == 64)
      gat_gemm_dual_small_wmma<64><<<ceil_div(N, 64), TPB, shmem, stream>>>(
          h1, Wl2, Wr2, xl2, xr2, N, H, Fout);
    else
      gat_gemm_dual_small_wmma<0><<<ceil_div(N, 64), TPB, shmem, stream>>>(
          h1, Wl2, Wr2, xl2, xr2, N, H, Fout);
  }

  if (Fout >= 32) {
    gat_edge_logits<<<ceil_div((long long)Etot * 32, TPB), TPB, 0, stream>>>(
        xl2, xr2, att2, srcE, dstE, ebuf, mord, E, Etot, Fout);
  } else {
    gat_edge_logits_small<<<ceil_div(Etot, TPB), TPB, 0, stream>>>(
        xl2, xr2, att2, srcE, dstE, ebuf, mord, E, Etot, Fout);
  }

  gat_edge_softmax_denom<<<ceil_div(Etot, TPB), TPB, 0, stream>>>(
      dstE, ebuf, mord, denom, E, Etot);

  if (Fout >= 32) {
    gat_edge_aggregate<<<ceil_div((long long)Etot * 32, TPB), TPB, 0, stream>>>(
        xl2, srcE, dstE, ebuf, denom, out, E, Etot, Fout);
  } else {
    gat_edge_aggregate_small<<<ceil_div(Etot, TPB), TPB, 0, stream>>>(
        xl2, srcE, dstE, ebuf, denom, out, E, Etot, Fout);
  }
}